// VectorQuantizer2_25744033972406
// MI455X (gfx1250) — compile-verified
//
#include <hip/hip_runtime.h>

// MI455X / gfx1250, wave32. f32 WMMA (16x16x4) for codebook GEMM + Phi conv;
// codebook tiles staged to LDS via async global->LDS DMA (ASYNCcnt) to kill
// the L2 re-read bottleneck; conv weights staged to LDS once per block.

typedef __attribute__((ext_vector_type(2))) float v2f;
typedef __attribute__((ext_vector_type(8))) float v8f;

#define Bq 1024
#define Cc 64
#define Hh 32
#define Vv 4096
#define ELEMS (Bq * Cc * Hh)   // 2,097,152
#define VCHUNK 64              // codes staged per LDS buffer (16 KB)

// torch bicubic kernel, a = -0.75 (matches reference _cubic)
__device__ __forceinline__ float cubicw(float x) {
  x = fabsf(x);
  const float a = -0.75f;
  if (x <= 1.f) return ((a + 2.f) * x - (a + 3.f)) * x * x + 1.f;
  if (x < 2.f)  return a * (((x - 5.f) * x + 8.f) * x - 4.f);
  return 0.f;
}

// ---------------------------------------------------------------- init
__global__ void init_state(const float* __restrict__ f,
                           float* __restrict__ f_hat,
                           float* __restrict__ f_rest) {
  int i = blockIdx.x * blockDim.x + threadIdx.x;
  if (i < ELEMS) { f_hat[i] = 0.f; f_rest[i] = f[i]; }
}

// per-code squared norms; also zero the loss accumulators
__global__ void emb_sq(const float* __restrict__ emb,
                       float* __restrict__ esq,
                       float* __restrict__ lacc) {
  int v = blockIdx.x * blockDim.x + threadIdx.x;
  if (v < 16) lacc[v] = 0.f;
  if (v < Vv) {
    float s = 0.f;
    #pragma unroll 8
    for (int c = 0; c < Cc; ++c) { float e = emb[v * Cc + c]; s += e * e; }
    esq[v] = s;
  }
}

// ------------------------------------------- bicubic downsample H->ph (align_corners=True)
// writes rest row-major [b*ph + o, c]
__global__ void downsample(const float* __restrict__ src,
                           float* __restrict__ rest, int ph) {
  int i = blockIdx.x * blockDim.x + threadIdx.x;  // over B*ph*C
  int tot = Bq * ph * Cc;
  if (i >= tot) return;
  int c = i & (Cc - 1);
  int o = (i / Cc) % ph;
  int b = i / (Cc * ph);
  float s = (ph == 1) ? 0.f : (float)o * (float)(Hh - 1) / (float)(ph - 1);
  int fl = (int)floorf(s);
  float t = s - (float)fl;
  const float* x = src + ((size_t)b * Cc + c) * Hh;
  float acc = 0.f;
  #pragma unroll
  for (int k = -1; k <= 2; ++k) {
    int j = fl + k; j = j < 0 ? 0 : (j > Hh - 1 ? Hh - 1 : j);
    acc += cubicw((float)k - t) * x[j];
  }
  rest[(size_t)(b * ph + o) * Cc + c] = acc;
}

// last scale: rest_NC = f transposed to [b*H + h, c]
__global__ void transpose_last(const float* __restrict__ f,
                               float* __restrict__ rest) {
  int i = blockIdx.x * blockDim.x + threadIdx.x;
  if (i >= ELEMS) return;
  int c = i & (Cc - 1);
  int h = (i / Cc) & (Hh - 1);
  int b = i / (Cc * Hh);
  rest[i] = f[((size_t)b * Cc + c) * Hh + h];
}

// ------------------------------------------- WMMA codebook argmin
// 8 waves / block, 16 rows per wave (128 rows / block). Codebook streamed
// through LDS in 64-code chunks with double-buffered async global->LDS DMA.
// score = ||e||^2 - 2*(rest.e)  (row-constant ||r||^2 dropped: argmin-invariant)
__global__ __launch_bounds__(256) void argmin_codes(
    const float* __restrict__ rest, const float* __restrict__ emb,
    const float* __restrict__ esq, int* __restrict__ idx) {
  __shared__ float btile[2][VCHUNK * Cc];   // 2 x 16 KB
  const int tid  = threadIdx.x;
  const int lane = tid & 31;
  const int wv   = tid >> 5;
  const int m0   = blockIdx.x * 128 + wv * 16;
  const int r    = m0 + (lane & 15);
  const int kk   = (lane >> 4) * 2;         // K lane split: {0,1} / {2,3}

  // A fragment: whole 16x64 row block resident in 32 VGPRs per wave
  v2f af[16];
  #pragma unroll
  for (int kc = 0; kc < 16; ++kc)
    af[kc] = *(const v2f*)(rest + (size_t)r * Cc + kc * 4 + kk);

  float bs[8]; int bi[8];
  #pragma unroll
  for (int i = 0; i < 8; ++i) { bs[i] = 3.4e38f; bi[i] = 0; }

  const unsigned lds0 = (unsigned)(size_t)(&btile[0][0]);
  const uint64_t gemb = (uint64_t)(uintptr_t)emb;

  // prologue: stage chunk 0 into buffer 0 (256 threads x 4 x 16B = 16 KB)
  {
    unsigned lo = lds0 + (unsigned)tid * 16u;
    uint64_t g  = gemb + (uint64_t)tid * 16u;
    #pragma unroll
    for (int j = 0; j < 4; ++j)
      asm volatile("global_load_async_to_lds_b128 %0, %1, off"
                   :: "v"(lo + (unsigned)j * 4096u), "v"(g + (uint64_t)j * 4096u)
                   : "memory");
  }

  const int NCHUNK = Vv / VCHUNK;           // 64
  for (int t = 0; t < NCHUNK; ++t) {
    asm volatile("s_wait_asynccnt 0" ::: "memory");  // my part of chunk t landed
    __syncthreads();                                  // everyone's landed; prev buf free
    const int buf = t & 1;
    if (t + 1 < NCHUNK) {                             // kick off chunk t+1 -> other buf
      unsigned lo = lds0 + (unsigned)(buf ^ 1) * (VCHUNK * Cc * 4u) + (unsigned)tid * 16u;
      uint64_t g  = gemb + (uint64_t)(t + 1) * (VCHUNK * Cc * 4u) + (uint64_t)tid * 16u;
      #pragma unroll
      for (int j = 0; j < 4; ++j)
        asm volatile("global_load_async_to_lds_b128 %0, %1, off"
                     :: "v"(lo + (unsigned)j * 4096u), "v"(g + (uint64_t)j * 4096u)
                     : "memory");
    }
    const float* bt = &btile[buf][0];
    #pragma unroll
    for (int sub = 0; sub < VCHUNK / 16; ++sub) {
      const int cloc = sub * 16 + (lane & 15);
      const int col  = t * VCHUNK + cloc;
      v8f c = {};
      #pragma unroll
      for (int kc = 0; kc < 16; ++kc) {
        v2f bf = *(const v2f*)(bt + (size_t)cloc * Cc + kc * 4 + kk);  // ds_load_b64
        c = __builtin_amdgcn_wmma_f32_16x16x4_f32(false, af[kc], false, bf,
                                                  (short)0, c, false, false);
      }
      const float e2 = esq[col];
      #pragma unroll
      for (int i = 0; i < 8; ++i) {
        float s = e2 - 2.f * c[i];
        if (s < bs[i]) { bs[i] = s; bi[i] = col; }   // strict <: keep lowest idx
      }
    }
  }

  // butterfly argmin across the 16 lanes sharing each row half (wave32 shfl)
  #pragma unroll
  for (int off = 1; off <= 8; off <<= 1) {
    #pragma unroll
    for (int i = 0; i < 8; ++i) {
      float os = __shfl_xor(bs[i], off, 32);
      int   oi = __shfl_xor(bi[i], off, 32);
      if (os < bs[i] || (os == bs[i] && oi < bi[i])) { bs[i] = os; bi[i] = oi; }
    }
  }
  if ((lane & 15) == 0) {
    int base = m0 + ((lane >> 4) * 8);
    #pragma unroll
    for (int i = 0; i < 8; ++i) idx[base + i] = bi[i];
  }
}

// ------------------------------------------- gather codes + bicubic upsample ph->H
// (align_corners=False). Writes hbuf in f layout [b, c, h].
__global__ void gather_upsample(const int* __restrict__ idx,
                                const float* __restrict__ emb,
                                float* __restrict__ hbuf, int ph, int last) {
  int i = blockIdx.x * blockDim.x + threadIdx.x;
  if (i >= ELEMS) return;
  int h = i & (Hh - 1);
  int c = (i / Hh) & (Cc - 1);
  int b = i / (Hh * Cc);
  float val;
  if (last) {
    val = emb[(size_t)idx[b * Hh + h] * Cc + c];
  } else {
    float s = ((float)h + 0.5f) * (float)ph / (float)Hh - 0.5f;
    int fl = (int)floorf(s);
    float t = s - (float)fl;
    val = 0.f;
    #pragma unroll
    for (int k = -1; k <= 2; ++k) {
      int j = fl + k; j = j < 0 ? 0 : (j > ph - 1 ? ph - 1 : j);
      val += cubicw((float)k - t) * emb[(size_t)idx[b * ph + j] * Cc + c];
    }
  }
  hbuf[i] = val;
}

// ------------------------------------------- WMMA Phi conv (W=1 -> kw=1 only) + fused update
// 8 waves / block, each wave: 16 (b,h) rows x all 64 co (4 co tiles, A reused).
// Center-column weights [co][ci][kh] staged to LDS once per block (48 KB).
__global__ __launch_bounds__(256) void phi_update(
    const float* __restrict__ hbuf, const float* __restrict__ w,
    const float* __restrict__ bias, const float* __restrict__ f,
    float* __restrict__ f_hat, float* __restrict__ f_rest,
    float* __restrict__ level, float* __restrict__ fst,
    float* __restrict__ loss_acc) {
  __shared__ float wl[Cc * Cc * 3];          // [co][ci][kh]
  const int tid = threadIdx.x, lane = tid & 31, wv = tid >> 5;
  for (int t = tid; t < Cc * Cc * 3; t += 256) {
    int co = t / (Cc * 3);
    int rem = t - co * (Cc * 3);
    int ci = rem / 3, kh = rem - ci * 3;
    wl[t] = w[(size_t)co * (Cc * 9) + (size_t)ci * 9 + kh * 3 + 1];
  }
  __syncthreads();

  const int m0 = blockIdx.x * 128 + wv * 16;
  const int mrow = m0 + (lane & 15);
  const int bb = mrow >> 5, hh0 = mrow & 31;
  const int cbase = lane & 15;
  const int kk = (lane >> 4) * 2;

  v8f acc[4] = {{}, {}, {}, {}};
  #pragma unroll
  for (int kh = 0; kh < 3; ++kh) {
    int hh = hh0 + kh - 1;
    bool ok = (hh >= 0) && (hh < Hh);
    const float* xb = hbuf + (size_t)bb * (Cc * Hh) + (ok ? hh : 0);
    #pragma unroll
    for (int kc = 0; kc < 16; ++kc) {
      int ci = kc * 4 + kk;
      v2f a;
      a.x = ok ? xb[(size_t)ci * Hh] : 0.f;
      a.y = ok ? xb[(size_t)(ci + 1) * Hh] : 0.f;
      #pragma unroll
      for (int n = 0; n < 4; ++n) {
        int co = n * 16 + cbase;
        v2f bf;
        bf.x = wl[co * (Cc * 3) + ci * 3 + kh];
        bf.y = wl[co * (Cc * 3) + (ci + 1) * 3 + kh];
        acc[n] = __builtin_amdgcn_wmma_f32_16x16x4_f32(false, a, false, bf,
                                                       (short)0, acc[n], false, false);
      }
    }
  }

  float lsum = 0.f;
  const float rr = 0.5f;   // QRESI
  #pragma unroll
  for (int n = 0; n < 4; ++n) {
    int co = n * 16 + cbase;
    float bsv = bias[co];
    #pragma unroll
    for (int i = 0; i < 8; ++i) {
      int m = m0 + i + ((lane >> 4) * 8);
      int b2 = m >> 5, h2 = m & 31;
      size_t off = ((size_t)b2 * Cc + co) * Hh + h2;
      float y = acc[n][i] + bsv;
      float out = hbuf[off] * (1.f - rr) + y * rr;
      float fh = f_hat[off] + out;
      f_hat[off] = fh;
      f_rest[off] -= out;
      level[off] = fh;
      if (fst) fst[off] = fh;   // last scale: f_hat_st forward value == f_hat
      float d = fh - f[off];
      lsum += d * d;
    }
  }
  #pragma unroll
  for (int off = 1; off < 32; off <<= 1) lsum += __shfl_xor(lsum, off, 32);
  if (lane == 0) atomicAdd(loss_acc, lsum);
}

// loss = sum_si 1.25 * SSE_si / ELEMS / 10
__global__ void finish_loss(const float* __restrict__ acc, float* __restrict__ out) {
  if (threadIdx.x == 0 && blockIdx.x == 0) {
    float s = 0.f;
    for (int i = 0; i < 10; ++i) s += acc[i];
    out[0] = s * (1.25f / ((float)ELEMS * 10.f));
  }
}

extern "C" void kernel_launch(void* const* d_in, const int* in_sizes, int n_in,
                              void* d_out, int out_size, void* d_ws, size_t ws_size,
                              hipStream_t stream) {
  const float* f    = (const float*)d_in[0];  // [1024,64,32,1]
  const float* emb  = (const float*)d_in[1];  // [4096,64]
  const float* phiw = (const float*)d_in[2];  // [4,64,64,3,3]
  const float* phib = (const float*)d_in[3];  // [4,64]
  float* out = (float*)d_out;                 // f_hat_st | loss | levels[10]

  float* f_hat  = (float*)d_ws;
  float* f_rest = f_hat + ELEMS;
  float* rest   = f_rest + ELEMS;
  float* hbuf   = rest + ELEMS;
  float* esq    = hbuf + ELEMS;
  float* lacc   = esq + Vv;
  int*   idx    = (int*)(lacc + 16);          // up to 32768 ints

  static const int vpatch[10] = {1, 2, 3, 4, 6, 9, 13, 18, 24, 32};
  static const int phisel[10] = {0, 0, 0, 1, 1, 2, 2, 2, 3, 3};  // argmin|TICKS - si/9|

  init_state<<<ELEMS / 256, 256, 0, stream>>>(f, f_hat, f_rest);
  emb_sq<<<Vv / 256, 256, 0, stream>>>(emb, esq, lacc);

  float* levels = out + ELEMS + 1;
  for (int si = 0; si < 10; ++si) {
    int ph = vpatch[si];
    int N = Bq * ph;
    if (si != 9) {
      int tot = N * Cc;
      downsample<<<(tot + 255) / 256, 256, 0, stream>>>(f_rest, rest, ph);
    } else {
      transpose_last<<<ELEMS / 256, 256, 0, stream>>>(f, rest);
    }
    argmin_codes<<<N / 128, 256, 0, stream>>>(rest, emb, esq, idx);
    gather_upsample<<<ELEMS / 256, 256, 0, stream>>>(idx, emb, hbuf, ph, si == 9);
    int pi = phisel[si];
    phi_update<<<(Bq * Hh) / 128, 256, 0, stream>>>(
        hbuf, phiw + (size_t)pi * Cc * Cc * 9, phib + (size_t)pi * Cc,
        f, f_hat, f_rest, levels + (size_t)si * ELEMS,
        (si == 9) ? out : (float*)nullptr, lacc + si);
  }
  finish_loss<<<1, 32, 0, stream>>>(lacc, out + ELEMS);
}